// bunch_conv_sc_1_8976481648956
// MI455X (gfx1250) — compile-verified
//
#include <hip/hip_runtime.h>

typedef __attribute__((ext_vector_type(2))) float v2f;
typedef __attribute__((ext_vector_type(8))) float v8f;

#define N0 4096
#define N1 8192

// ---------------------------------------------------------------------------
// Prep: fold the per-node einsum weights into the feature vectors.
//   u0 = x0 @ W0[:,:,0]            [N0,16]
//   v0 = (inv_d1 * x0) @ W1[:,:,0] [N0,16]
//   u1 = x1 @ W0[:,:,1]            [N1,16]
//   v1 = x1 @ W1[:,:,1]            [N1,16]
// ---------------------------------------------------------------------------
__global__ __launch_bounds__(256) void prep_kernel(
    const float* __restrict__ x0, const float* __restrict__ x1,
    const float* __restrict__ d1, const float* __restrict__ W0,
    const float* __restrict__ W1, float* __restrict__ u0,
    float* __restrict__ v0, float* __restrict__ u1, float* __restrict__ v1) {
  const int idx = blockIdx.x * blockDim.x + threadIdx.x;  // over N1*16
  const int n = idx >> 4;
  const int o = idx & 15;
  if (n < N0) {
    const float* xr = x0 + n * 16;
    float su = 0.f, sv = 0.f;
#pragma unroll
    for (int i = 0; i < 16; ++i) {
      const float xi = xr[i];
      su += xi * W0[i * 32 + o * 2 + 0];
      sv += xi * W1[i * 32 + o * 2 + 0];
    }
    u0[idx] = su;
    v0[idx] = sv / d1[n];
  }
  {
    const float* xr = x1 + n * 16;
    float su = 0.f, sv = 0.f;
#pragma unroll
    for (int i = 0; i < 16; ++i) {
      const float xi = xr[i];
      su += xi * W0[i * 32 + o * 2 + 1];
      sv += xi * W1[i * 32 + o * 2 + 1];
    }
    u1[idx] = su;
    v1[idx] = sv;
  }
}

// ---------------------------------------------------------------------------
// y0 = relu( L0 @ u0  +  inv_d1 (.) (B1 @ u1) )      one 16-row tile / block
// 8 wave32 per block, K split 8 ways, LDS cross-wave reduction.
// A layout (f32 16x4): lanes 0-15 -> M=lane, VGPR{0,1}=K{0,1};
//                      lanes 16-31 -> M=lane-16, VGPR{0,1}=K{2,3}.
// B layout (f32 4x16): lanes 0-15 -> N=lane, VGPR{0,1}=K{0,1};
//                      lanes 16-31 -> N=lane-16, VGPR{0,1}=K{2,3}.
// C layout: VGPR r, lanes 0-15 -> M=r, N=lane; lanes 16-31 -> M=r+8.
// ---------------------------------------------------------------------------
__global__ __launch_bounds__(256) void y0_kernel(
    const float* __restrict__ L0, const float* __restrict__ B1,
    const float* __restrict__ u0, const float* __restrict__ u1,
    const float* __restrict__ d1, float* __restrict__ out) {
  __shared__ float red[8][8][32];
  const int m0 = blockIdx.x * 16;
  const int lane = threadIdx.x & 31;
  const int wave = threadIdx.x >> 5;
  const int mrow = lane & 15;
  const int khalf = (lane >> 4) << 1;  // 0 or 2

  v8f accL = {};
  v8f accB = {};

  {  // L0 @ u0, this wave's K slice
    const float* Arow = L0 + (size_t)(m0 + mrow) * N0 + khalf;
    const float* Bp = u0 + khalf * 16 + mrow;
    const int k0 = wave * (N0 / 8);
#pragma unroll 4
    for (int k = k0; k < k0 + N0 / 8; k += 4) {
      v2f a = *(const v2f*)(Arow + k);
      v2f b;
      b.x = Bp[k * 16];
      b.y = Bp[k * 16 + 16];
      accL = __builtin_amdgcn_wmma_f32_16x16x4_f32(false, a, false, b,
                                                   (short)0, accL, false, false);
    }
  }
  {  // B1 @ u1, this wave's K slice
    const float* Arow = B1 + (size_t)(m0 + mrow) * N1 + khalf;
    const float* Bp = u1 + khalf * 16 + mrow;
    const int k0 = wave * (N1 / 8);
#pragma unroll 4
    for (int k = k0; k < k0 + N1 / 8; k += 4) {
      v2f a = *(const v2f*)(Arow + k);
      v2f b;
      b.x = Bp[k * 16];
      b.y = Bp[k * 16 + 16];
      accB = __builtin_amdgcn_wmma_f32_16x16x4_f32(false, a, false, b,
                                                   (short)0, accB, false, false);
    }
  }

  const int mhi = (lane >> 4) * 8;
#pragma unroll
  for (int r = 0; r < 8; ++r) {
    const int m = m0 + mhi + r;
    red[wave][r][lane] = accL[r] + accB[r] / d1[m];
  }
  __syncthreads();
  if (wave == 0) {
#pragma unroll
    for (int r = 0; r < 8; ++r) {
      float c = 0.f;
#pragma unroll
      for (int w = 0; w < 8; ++w) c += red[w][r][lane];
      const int m = m0 + mhi + r;
      out[m * 16 + mrow] = fmaxf(c, 0.f);
    }
  }
}

// ---------------------------------------------------------------------------
// y1 = relu( d2 (.) (B1^T @ v0)  +  L1 @ v1 )
// B1^T tile: element (m,k) = B1[k*N1 + m] -> 64B-contiguous column reads.
// ---------------------------------------------------------------------------
__global__ __launch_bounds__(256) void y1_kernel(
    const float* __restrict__ L1, const float* __restrict__ B1,
    const float* __restrict__ v0, const float* __restrict__ v1,
    const float* __restrict__ d2, float* __restrict__ out) {
  __shared__ float red[8][8][32];
  const int m0 = blockIdx.x * 16;
  const int lane = threadIdx.x & 31;
  const int wave = threadIdx.x >> 5;
  const int mrow = lane & 15;
  const int khalf = (lane >> 4) << 1;

  v8f accL = {};
  v8f accT = {};

  {  // L1 @ v1
    const float* Arow = L1 + (size_t)(m0 + mrow) * N1 + khalf;
    const float* Bp = v1 + khalf * 16 + mrow;
    const int k0 = wave * (N1 / 8);
#pragma unroll 4
    for (int k = k0; k < k0 + N1 / 8; k += 4) {
      v2f a = *(const v2f*)(Arow + k);
      v2f b;
      b.x = Bp[k * 16];
      b.y = Bp[k * 16 + 16];
      accL = __builtin_amdgcn_wmma_f32_16x16x4_f32(false, a, false, b,
                                                   (short)0, accL, false, false);
    }
  }
  {  // B1^T @ v0
    const float* Acol = B1 + (size_t)khalf * N1 + (m0 + mrow);
    const float* Bp = v0 + khalf * 16 + mrow;
    const int k0 = wave * (N0 / 8);
#pragma unroll 4
    for (int k = k0; k < k0 + N0 / 8; k += 4) {
      v2f a;
      a.x = Acol[(size_t)k * N1];
      a.y = Acol[(size_t)k * N1 + N1];
      v2f b;
      b.x = Bp[k * 16];
      b.y = Bp[k * 16 + 16];
      accT = __builtin_amdgcn_wmma_f32_16x16x4_f32(false, a, false, b,
                                                   (short)0, accT, false, false);
    }
  }

  const int mhi = (lane >> 4) * 8;
#pragma unroll
  for (int r = 0; r < 8; ++r) {
    const int m = m0 + mhi + r;
    red[wave][r][lane] = accL[r] + d2[m] * accT[r];
  }
  __syncthreads();
  if (wave == 0) {
#pragma unroll
    for (int r = 0; r < 8; ++r) {
      float c = 0.f;
#pragma unroll
      for (int w = 0; w < 8; ++w) c += red[w][r][lane];
      const int m = m0 + mhi + r;
      out[m * 16 + mrow] = fmaxf(c, 0.f);
    }
  }
}

extern "C" void kernel_launch(void* const* d_in, const int* in_sizes, int n_in,
                              void* d_out, int out_size, void* d_ws, size_t ws_size,
                              hipStream_t stream) {
  const float* x0 = (const float*)d_in[0];
  const float* x1 = (const float*)d_in[1];
  const float* L0 = (const float*)d_in[2];
  const float* L1 = (const float*)d_in[3];
  const float* B1 = (const float*)d_in[4];
  const float* d1 = (const float*)d_in[5];
  const float* d2 = (const float*)d_in[6];
  const float* W0 = (const float*)d_in[7];
  const float* W1 = (const float*)d_in[8];
  float* out = (float*)d_out;

  float* ws = (float*)d_ws;
  float* u0 = ws;                      // N0*16
  float* v0 = ws + N0 * 16;            // N0*16
  float* u1 = ws + 2 * N0 * 16;        // N1*16
  float* v1 = ws + 2 * N0 * 16 + N1 * 16;  // N1*16

  prep_kernel<<<(N1 * 16) / 256, 256, 0, stream>>>(x0, x1, d1, W0, W1, u0, v0,
                                                   u1, v1);
  y0_kernel<<<N0 / 16, 256, 0, stream>>>(L0, B1, u0, u1, d1, out);
  y1_kernel<<<N1 / 16, 256, 0, stream>>>(L1, B1, v0, v1, d2, out + N0 * 16);
}